// SAGEConv_19731079758620
// MI455X (gfx1250) — compile-verified
//
#include <hip/hip_runtime.h>
#include <hip/hip_bf16.h>

// ---------------------------------------------------------------------------
// SAGEConv (mean aggregation):
//   out = (segment_mean(x[src] -> dst)) @ W_l^T + x @ W_r^T
// Inputs (fp32 / int64):
//   d_in[0] node_feature [N,128] f32
//   d_in[1] W_l          [128,128] f32
//   d_in[2] W_r          [128,128] f32
//   d_in[3] edge_index   [2,E] int64  (row 0 = src, row 1 = dst)
// Output: [N,128] f32
// Workspace: summed [N*128] f32, deg [N] f32
// ---------------------------------------------------------------------------

typedef __attribute__((ext_vector_type(2))) float v2f;
typedef __attribute__((ext_vector_type(8))) float v8f;

#define D_FEAT 128

// ---------------------------------------------------------------------------
// Kernel 1: zero the accumulators (harness poisons d_ws with 0xAA).
// ---------------------------------------------------------------------------
__global__ void sage_zero_ws(float* __restrict__ ws, long long total) {
  long long i = (long long)blockIdx.x * blockDim.x + threadIdx.x;
  long long stride = (long long)gridDim.x * blockDim.x;
  for (; i < total; i += stride) ws[i] = 0.0f;
}

// ---------------------------------------------------------------------------
// Kernel 2: edge scatter. One wave (32 lanes) per edge; each lane moves 4
// floats of the 128-wide feature. Edge id is wave-uniform: readfirstlane
// forces the int64 index fetches onto the scalar (SMEM) path. The f32 adds
// use the hardware global_atomic_add_f32 path (no CAS loop); the 51 MB
// accumulator and the 51 MB feature table are both L2-resident (192 MB L2).
// ---------------------------------------------------------------------------
__global__ void sage_scatter(const float* __restrict__ xfeat,
                             const long long* __restrict__ src,
                             const long long* __restrict__ dst,
                             float* __restrict__ summed,
                             float* __restrict__ deg,
                             int n_edges) {
  const int lane = threadIdx.x & 31;
  int e = blockIdx.x * (blockDim.x >> 5) + (threadIdx.x >> 5);
  if (e >= n_edges) return;          // wave-uniform guard
  e = __builtin_amdgcn_readfirstlane(e);

  const int s = (int)src[e];  // scalar loads (uniform address)
  const int d = (int)dst[e];

  const float4 v =
      *(const float4*)(xfeat + (size_t)s * D_FEAT + (size_t)lane * 4);
  float* o = summed + (size_t)d * D_FEAT + (size_t)lane * 4;
  unsafeAtomicAdd(o + 0, v.x);
  unsafeAtomicAdd(o + 1, v.y);
  unsafeAtomicAdd(o + 2, v.z);
  unsafeAtomicAdd(o + 3, v.w);

  if (lane == 0) unsafeAtomicAdd(deg + d, 1.0f);
}

// ---------------------------------------------------------------------------
// Kernel 3: dual GEMM with V_WMMA_F32_16X16X4_F32 (full fp32 precision).
//   out[m,n] = invdeg[m] * (summed @ W_l^T)[m,n] + (x @ W_r^T)[m,n]
// The per-row mean divide commutes out of the GEMM and is applied in the
// epilogue, keeping the WMMA dependency chain free of VALU ops.
//
// Block = 256 threads = 8 waves; block covers 32 output rows.
// Wave w computes two 16x16 tiles: rows [m0,m0+16) and [m0+16,m0+32),
// cols [16w,16w+16). Each B fragment (weights) feeds 2 WMMAs.
//
// f32 16x16x4 VGPR layout (ISA 7.12.2):
//   A: lane holds float2 of row m0+(lane&15), ks = k + 2*(lane>>4)
//   B: lane holds float2 of col n0+(lane&15): B[k][n] = W[n][k] -> contiguous
//   C/D: acc[v] -> out[m0 + v + 8*(lane>>4)][n0 + (lane&15)]
// ---------------------------------------------------------------------------
__global__ void __launch_bounds__(256)
sage_gemm_wmma(const float* __restrict__ xfeat,
               const float* __restrict__ Wl,
               const float* __restrict__ Wr,
               const float* __restrict__ summed,
               const float* __restrict__ deg,
               float* __restrict__ out,
               int n_nodes) {
  const int lane = threadIdx.x & 31;
  const int wave = threadIdx.x >> 5;  // 0..7 -> N tile
  const int m0 = blockIdx.x * 32;
  const int n0 = wave * 16;

  const int r = lane & 15;         // A row-in-tile / B col-in-tile
  const int kh = (lane >> 4) * 2;  // K sub-offset: 0 or 2

  int row0 = m0 + r;
  int row1 = m0 + 16 + r;
  if (row0 >= n_nodes) row0 = n_nodes - 1;  // clamp loads for ragged tail
  if (row1 >= n_nodes) row1 = n_nodes - 1;

  const float* __restrict__ as0 = summed + (size_t)row0 * D_FEAT + kh;
  const float* __restrict__ as1 = summed + (size_t)row1 * D_FEAT + kh;
  const float* __restrict__ ax0 = xfeat + (size_t)row0 * D_FEAT + kh;
  const float* __restrict__ ax1 = xfeat + (size_t)row1 * D_FEAT + kh;
  const float* __restrict__ bl = Wl + (size_t)(n0 + r) * D_FEAT + kh;
  const float* __restrict__ br = Wr + (size_t)(n0 + r) * D_FEAT + kh;

  v8f accl0 = {}, accl1 = {};  // summed @ W_l^T, two M tiles
  v8f accr0 = {}, accr1 = {};  // x @ W_r^T,      two M tiles

  // summed @ W_l^T : one B fragment shared by two WMMAs
#pragma unroll 4
  for (int k = 0; k < D_FEAT; k += 4) {
    const v2f b = *(const v2f*)(bl + k);
    const v2f a0 = *(const v2f*)(as0 + k);
    const v2f a1 = *(const v2f*)(as1 + k);
    accl0 = __builtin_amdgcn_wmma_f32_16x16x4_f32(false, a0, false, b,
                                                  (short)0, accl0, false, false);
    accl1 = __builtin_amdgcn_wmma_f32_16x16x4_f32(false, a1, false, b,
                                                  (short)0, accl1, false, false);
  }

  // x @ W_r^T
#pragma unroll 4
  for (int k = 0; k < D_FEAT; k += 4) {
    const v2f b = *(const v2f*)(br + k);
    const v2f a0 = *(const v2f*)(ax0 + k);
    const v2f a1 = *(const v2f*)(ax1 + k);
    accr0 = __builtin_amdgcn_wmma_f32_16x16x4_f32(false, a0, false, b,
                                                  (short)0, accr0, false, false);
    accr1 = __builtin_amdgcn_wmma_f32_16x16x4_f32(false, a1, false, b,
                                                  (short)0, accr1, false, false);
  }

  // Epilogue: apply the mean (per output row) and combine the two GEMMs.
  const int half = (lane >> 4) * 8;
#pragma unroll
  for (int v = 0; v < 8; ++v) {
    const int ra = m0 + half + v;
    if (ra < n_nodes) {
      const float invd = 1.0f / fmaxf(deg[ra], 1.0f);
      out[(size_t)ra * D_FEAT + n0 + r] = invd * accl0[v] + accr0[v];
    }
    const int rb = m0 + 16 + half + v;
    if (rb < n_nodes) {
      const float invd = 1.0f / fmaxf(deg[rb], 1.0f);
      out[(size_t)rb * D_FEAT + n0 + r] = invd * accl1[v] + accr1[v];
    }
  }
}

// ---------------------------------------------------------------------------
extern "C" void kernel_launch(void* const* d_in, const int* in_sizes, int n_in,
                              void* d_out, int out_size, void* d_ws,
                              size_t ws_size, hipStream_t stream) {
  const float* xfeat = (const float*)d_in[0];
  const float* Wl = (const float*)d_in[1];
  const float* Wr = (const float*)d_in[2];
  const long long* edge = (const long long*)d_in[3];

  const int n_nodes = in_sizes[0] / D_FEAT;
  const int n_edges = in_sizes[3] / 2;

  const long long* src = edge;
  const long long* dst = edge + n_edges;

  float* summed = (float*)d_ws;                    // [n_nodes * 128]
  float* deg = summed + (size_t)n_nodes * D_FEAT;  // [n_nodes]
  float* outp = (float*)d_out;

  // 1) zero accumulators
  {
    const long long total = (long long)n_nodes * D_FEAT + n_nodes;
    int blocks = (int)((total + 255) / 256);
    if (blocks > 65535) blocks = 65535;
    sage_zero_ws<<<blocks, 256, 0, stream>>>(summed, total);
  }

  // 2) edge scatter (mean numerator + degree)
  {
    const int waves_per_block = 8;  // 256 threads
    const int blocks = (n_edges + waves_per_block - 1) / waves_per_block;
    sage_scatter<<<blocks, 256, 0, stream>>>(xfeat, src, dst, summed, deg,
                                             n_edges);
  }

  // 3) dual GEMM (fp32 WMMA) + fused mean epilogue
  {
    const int blocks = (n_nodes + 31) / 32;  // 3125 for N=100000
    sage_gemm_wmma<<<blocks, 256, 0, stream>>>(xfeat, Wl, Wr, summed, deg,
                                               outp, n_nodes);
  }
}